// Beran_75161927680288
// MI455X (gfx1250) — compile-verified
//
#include <hip/hip_runtime.h>
#include <hip/hip_bf16.h>
#include <math.h>

typedef __attribute__((ext_vector_type(2))) float v2f;
typedef __attribute__((ext_vector_type(8))) float v8f;

#define BATCH 128
#define NPTS  4096
#define DDIM  256
#define EPS_SUM 1e-13f
#define ISCLOSE_TOL 1.00001e-5f   // atol + rtol*|1.0| = 1e-8 + 1e-5

// ---------------------------------------------------------------------------
// Kernel 1: streaming distance + kernel-weight pass (bandwidth-bound, 536 MB).
// ||x - p||^2 = x.x - 2 x.p + p.p ; the x.p term for a 16-row tile is computed
// with V_WMMA_F32_16X16X4_F32, p replicated across all 16 B-columns.
// One block = 128 rows of batch b (8 waves => 8 tiles of 16 rows); p staged in LDS.
// ---------------------------------------------------------------------------
__global__ __launch_bounds__(256)
void beran_weights_kernel(const float* __restrict__ x_in,
                          const float* __restrict__ x_p,
                          const float* __restrict__ bandwidth,
                          float* __restrict__ w_out) {
    __shared__ float p_lds[DDIM];

    const int tid  = threadIdx.x;
    const int lane = tid & 31;
    const int wave = tid >> 5;
    const int tileGroup = blockIdx.x & 31;   // 32 tile-groups per batch
    const int b         = blockIdx.x >> 5;

    // Stage p row (256 floats) into LDS: one float per thread.
    p_lds[tid] = x_p[b * DDIM + tid];
    __syncthreads();

    const int i0  = (tileGroup * 8 + wave) * 16;  // first n-row of this tile
    const int row = lane & 15;                    // A-matrix M index
    const int hi  = lane >> 4;                    // lane half selects K pair
    const float* xrow_ptr =
        x_in + ((size_t)b * NPTS + i0 + row) * DDIM + 2 * hi;

    // p.p (wave-uniform): strided square-sum + xor-shuffle reduction.
    float psq = 0.f;
    #pragma unroll
    for (int k = lane; k < DDIM; k += 32) {
        float pv = p_lds[k];
        psq += pv * pv;
    }
    #pragma unroll
    for (int off = 16; off > 0; off >>= 1) psq += __shfl_xor(psq, off);

    // Accumulate x.p via WMMA and x.x per lane over K.
    v8f c = {0.f, 0.f, 0.f, 0.f, 0.f, 0.f, 0.f, 0.f};
    float sq = 0.f;
    #pragma unroll 4
    for (int k0 = 0; k0 < DDIM; k0 += 4) {
        if ((k0 & 63) == 0)
            __builtin_prefetch(xrow_ptr + k0 + 64, 0, 0);  // global_prefetch_b8
        v2f a = *(const v2f*)(xrow_ptr + k0);              // A: row, K=2hi+{0,1}
        v2f bm = { p_lds[k0 + 2 * hi], p_lds[k0 + 2 * hi + 1] };  // B: p bcast
        sq += a.x * a.x + a.y * a.y;
        // D = A(16x4) * B(4x16) + C  ; every D column holds the row dot.
        c = __builtin_amdgcn_wmma_f32_16x16x4_f32(
                /*neg_a=*/false, a, /*neg_b=*/false, bm,
                /*c_mod=*/(short)0, c, /*reuse_a=*/false, /*reuse_b=*/false);
    }

    // x.x per row: lanes L and L^16 cover disjoint K halves of row L%16.
    float xsq = sq + __shfl_xor(sq, 16);

    float bw = bandwidth[0];
    bw = fminf(fmaxf(bw, 0.1f), 10.0f);

    union { v8f v; float f[8]; } cu;
    cu.v = c;

    // D VGPR r holds row r (lanes 0-15) and row r+8 (lanes 16-31).
    #pragma unroll
    for (int r = 0; r < 8; ++r) {
        int srcl   = r + ((lane & 16) >> 1);     // lane holding xsq of my row
        float xr   = __shfl(xsq, srcl);
        float dot  = cu.f[r];
        float d2   = fmaxf(xr - 2.0f * dot + psq, 0.0f);
        float met  = sqrtf(d2);
        float w    = __expf(-met / bw);
        if ((lane & 15) == 0) {                  // lanes 0 and 16 write
            int m = r + 8 * hi;
            w_out[(size_t)b * NPTS + i0 + m] = w;
        }
    }
}

// ---------------------------------------------------------------------------
// Kernel 2: per-batch normalization + two inclusive scans + survival outputs.
// One block per batch row; 256 threads x 16 contiguous elements each.
// ---------------------------------------------------------------------------
__global__ __launch_bounds__(256)
void beran_scan_kernel(const float* __restrict__ w_in,
                       const float* __restrict__ delta_in,
                       float* __restrict__ surv_out,    // [B, N]
                       float* __restrict__ steps_out) { // [B, N-1]
    __shared__ float surv_lds[NPTS];   // 16 KB
    __shared__ float tot[256];

    const int t = threadIdx.x;
    const int b = blockIdx.x;
    const int CH = NPTS / 256;         // 16
    const size_t base = (size_t)b * NPTS;

    float w[CH], loc[CH];
    #pragma unroll
    for (int j = 0; j < CH; ++j) w[j] = w_in[base + t * CH + j];

    // Local inclusive scan of raw weights.
    float run = 0.f;
    #pragma unroll
    for (int j = 0; j < CH; ++j) { run += w[j]; loc[j] = run; }
    tot[t] = run;
    __syncthreads();
    // Hillis-Steele inclusive scan over 256 thread totals.
    for (int off = 1; off < 256; off <<= 1) {
        float add = (t >= off) ? tot[t - off] : 0.f;
        __syncthreads();
        tot[t] += add;
        __syncthreads();
    }
    float S      = tot[255];
    float offset = tot[t] - run;                    // exclusive prefix
    float invS   = (S < EPS_SUM) ? 0.f : (1.0f / S);

    // xi and delta*xi per element.
    float tv[CH];
    #pragma unroll
    for (int j = 0; j < CH; ++j) {
        float cum     = (loc[j] + offset) * invS;   // cumsum of normalized W
        float Wn      = w[j] * invS;
        float shifted = cum - Wn;
        bool bad = (fabsf(shifted - 1.0f) <= ISCLOSE_TOL) ||
                   (fabsf(cum     - 1.0f) <= ISCLOSE_TOL);
        if (bad) { shifted = 0.f; cum = 0.f; }
        float xi  = logf(1.0f - shifted) - logf(1.0f - cum);
        float dlt = delta_in[base + t * CH + j];
        tv[j] = dlt * xi;
    }

    // Second scan: hazards = cumsum(delta * xi).
    run = 0.f;
    #pragma unroll
    for (int j = 0; j < CH; ++j) { run += tv[j]; tv[j] = run; }
    __syncthreads();
    tot[t] = run;
    __syncthreads();
    for (int off = 1; off < 256; off <<= 1) {
        float add = (t >= off) ? tot[t - off] : 0.f;
        __syncthreads();
        tot[t] += add;
        __syncthreads();
    }
    offset = tot[t] - run;

    #pragma unroll
    for (int j = 0; j < CH; ++j) {
        float hz = tv[j] + offset;
        float sv = __expf(-hz);
        surv_lds[t * CH + j]      = sv;
        surv_out[base + t * CH + j] = sv;
    }
    __syncthreads();

    const size_t base2 = (size_t)b * (NPTS - 1);
    #pragma unroll
    for (int j = 0; j < CH; ++j) {
        int idx = t * CH + j;
        if (idx < NPTS - 1)
            steps_out[base2 + idx] = surv_lds[idx] - surv_lds[idx + 1];
    }
}

// ---------------------------------------------------------------------------
extern "C" void kernel_launch(void* const* d_in, const int* in_sizes, int n_in,
                              void* d_out, int out_size, void* d_ws, size_t ws_size,
                              hipStream_t stream) {
    const float* x_in      = (const float*)d_in[0]; // [B, n, d] f32
    const float* delta_in  = (const float*)d_in[1]; // [B, n] f32
    const float* x_p       = (const float*)d_in[2]; // [B, d] f32
    const float* bandwidth = (const float*)d_in[3]; // [1] f32

    float* surv  = (float*)d_out;                       // [B, n]
    float* steps = surv + (size_t)BATCH * NPTS;         // [B, n-1]
    float* wbuf  = (float*)d_ws;                        // [B, n] scratch (2 MB)

    // Phase 1: 128 batches * 32 tile-groups = 4096 blocks of 256 threads.
    beran_weights_kernel<<<BATCH * 32, 256, 0, stream>>>(x_in, x_p, bandwidth, wbuf);
    // Phase 2: one block per batch row.
    beran_scan_kernel<<<BATCH, 256, 0, stream>>>(wbuf, delta_in, surv, steps);
}